// LIFCell_45354854646141
// MI455X (gfx1250) — compile-verified
//
#include <hip/hip_runtime.h>
#include <stdint.h>

// ---------------------------------------------------------------------------
// Fused LIF cell for MI455X (gfx1250).
//   I = M @ W^T  (T=64, N=50000, D=128)  then sequential LIF scan over T.
// Bandwidth bound: 1.64 GB of M at 23.3 TB/s ~= 70 us floor.
// Path: async-tensor copies (global_load_async_to_lds_b128, double buffered,
// s_wait_asynccnt) + v_wmma_f32_16x16x4_f32 with broadcast-W B matrix held
// entirely in VGPRs (loop-invariant), so the hot loop only DS-loads A tiles.
// ---------------------------------------------------------------------------

#define T_STEPS 64
#define NNEUR   50000
#define DDIM    128
#define WAVES_PER_BLOCK 8
#define ROW_F   132                    // padded LDS row stride (floats) = 528B
#define BUF_F   (16 * ROW_F)           // one 16x128 tile buffer (floats)
#define LDS_W_OFF (WAVES_PER_BLOCK * 2 * BUF_F)

typedef __attribute__((ext_vector_type(2))) float v2f;
typedef __attribute__((ext_vector_type(8))) float v8f;

// Async-copy one 16x128 f32 tile (global, row-major, row stride 128 floats)
// into LDS at byte offset lds_byte_base with 528B padded rows.
// One b128 instruction moves one full 512B row (32 lanes x 16B).
__device__ __forceinline__ void async_tile_load(const float* g,
                                                unsigned lds_byte_base,
                                                int lane) {
#pragma unroll
  for (int i = 0; i < 16; ++i) {
    const float* gp = g + i * DDIM + lane * 4;
    unsigned lp = lds_byte_base + (unsigned)(i * (ROW_F * 4) + lane * 16);
    asm volatile("global_load_async_to_lds_b128 %0, %1, off"
                 :: "v"(lp), "v"((unsigned long long)(uintptr_t)gp)
                 : "memory");
  }
}

__global__ __launch_bounds__(32 * WAVES_PER_BLOCK)
void lif_fused_kernel(const float* __restrict__ M,
                      const float* __restrict__ W,
                      float* __restrict__ out) {
  __shared__ float lds[LDS_W_OFF + DDIM];

  const int tid  = threadIdx.x;
  const int lane = tid & 31;
  const int wv   = tid >> 5;
  const int r    = lane & 15;   // A row-in-tile / B column index
  const int h    = lane >> 4;   // which K/M half this lane carries
  const int j    = lane & 7;    // accumulator element for this lane's neuron

  // Stage W (128 floats) once per block.
  if (tid < DDIM) lds[LDS_W_OFF + tid] = W[tid];

  const int  tile   = blockIdx.x * WAVES_PER_BLOCK + wv;
  const bool active = (tile * 16) < NNEUR;
  const int  n0     = tile * 16;

  const float*   Mbase  = M + (size_t)n0 * DDIM;
  float* const   bufGen = &lds[wv * 2 * BUF_F];
  const unsigned buf0   = (unsigned)(uintptr_t)(void*)bufGen;   // LDS byte offset
  const unsigned buf1   = buf0 + (unsigned)(BUF_F * 4);

  // Prefetch t = 0 before the barrier so it overlaps W staging.
  if (active) async_tile_load(Mbase, buf0, lane);
  __syncthreads();
  if (!active) return;

  const float* ldsWp = &lds[LDS_W_OFF];

  // Hoist the loop-invariant B operands (broadcast W) into VGPRs:
  // per lane, B pair for k-block `blk` is W[4*blk + 2h + {0,1}].
  v2f wreg[32];
#pragma unroll
  for (int blk = 0; blk < 32; ++blk) {
    wreg[blk] = *(const v2f*)(ldsWp + blk * 4 + 2 * h);
  }

  float* __restrict__ outS = out;
  float* __restrict__ outV = out + (size_t)T_STEPS * NNEUR;
  float* __restrict__ outT = out + 2 * (size_t)T_STEPS * NNEUR;

  const float LAM = 0.95f, TAU = 0.99f, GAM = 0.1f;
  float V = 0.f, Th = 1.f, Sp = 0.f;

  for (int t = 0; t < T_STEPS; ++t) {
    if (t + 1 < T_STEPS) {
      // Issue next tile, then wait only for the current tile (<=16 in flight).
      async_tile_load(Mbase + (size_t)(t + 1) * NNEUR * DDIM,
                      ((t & 1) ? buf0 : buf1), lane);
      asm volatile("s_wait_asynccnt 16" ::: "memory");
    } else {
      asm volatile("s_wait_asynccnt 0" ::: "memory");
    }

    const float* A = bufGen + (t & 1) * BUF_F;

    float Ival;
#if __has_builtin(__builtin_amdgcn_wmma_f32_16x16x4_f32)
    // D = A(16x128 slice) x B + C, K tiled by 4. B columns all equal W, so
    // every column of D is the I vector for these 16 neurons.
    // Per-lane A operand: A[r, 4*blk + 2h + {0,1}] ; B operand: wreg[blk].
    v8f c = {};
#pragma unroll
    for (int blk = 0; blk < 32; ++blk) {
      v2f a = *(const v2f*)(A + r * ROW_F + blk * 4 + 2 * h);
      c = __builtin_amdgcn_wmma_f32_16x16x4_f32(
              /*neg_a=*/false, a, /*neg_b=*/false, wreg[blk],
              /*c_mod=*/(short)0, c, /*reuse_a=*/false, /*reuse_b=*/false);
    }
    // Lane (h, j) handles neuron n0 + 8h + j; its I value is c[j].
    {
      float s01 = (j & 1) ? c[1] : c[0];
      float s23 = (j & 1) ? c[3] : c[2];
      float s45 = (j & 1) ? c[5] : c[4];
      float s67 = (j & 1) ? c[7] : c[6];
      float a03 = (j & 2) ? s23 : s01;
      float a47 = (j & 2) ? s67 : s45;
      Ival      = (j & 4) ? a47 : a03;
    }
#else
    // VALU fallback: direct dot product for this lane's neuron.
    {
      const float* row = A + (8 * h + j) * ROW_F;
      float acc = 0.f;
#pragma unroll
      for (int d = 0; d < DDIM; ++d) acc = __builtin_fmaf(row[d], ldsWp[d], acc);
      Ival = acc;
    }
#endif

    // LIF update (matches reference ordering; hard spike = STE forward).
    V = LAM * V + Ival - Th * Sp;
    float diff = V - Th;
    float S = (diff > 0.f) ? 1.f : 0.f;
    V = V - S * Th;
    float Tn = TAU * Th + GAM * Sp;

    if (r < 8) {  // lanes 0-7 -> neurons n0..n0+7, lanes 16-23 -> n0+8..n0+15
      size_t idx = (size_t)t * NNEUR + (size_t)(n0 + 8 * h + j);
      outS[idx] = S;
      outV[idx] = V;
      outT[idx] = Tn;
    }

    Sp = S;
    Th = Tn;
  }
}

extern "C" void kernel_launch(void* const* d_in, const int* in_sizes, int n_in,
                              void* d_out, int out_size, void* d_ws, size_t ws_size,
                              hipStream_t stream) {
  (void)in_sizes; (void)n_in; (void)d_ws; (void)ws_size; (void)out_size;
  const float* M = (const float*)d_in[0];   // [T, N, D] f32
  const float* W = (const float*)d_in[1];   // [1, D]    f32
  float* out = (float*)d_out;               // [3, T, N] f32 (S, V, theta)

  const int tiles  = (NNEUR + 15) / 16;                               // 3125
  const int blocks = (tiles + WAVES_PER_BLOCK - 1) / WAVES_PER_BLOCK; // 391
  lif_fused_kernel<<<blocks, 32 * WAVES_PER_BLOCK, 0, stream>>>(M, W, out);
}